// GIN_72507637891671
// MI455X (gfx1250) — compile-verified
//
#include <hip/hip_runtime.h>
#include <hip/hip_bf16.h>

typedef __attribute__((ext_vector_type(2))) float v2f;
typedef __attribute__((ext_vector_type(8))) float v8f;

#define N_NODES 100000
#define N_EDGES 1600000
#define IN_DIM 128
#define HID 64
#define OUT_DIM 10
#define N_GRAPHS 512
#define BN_EPS 1e-5f

// ---------------------------------------------------------------------------
// Scatter-add: z[dst[e]*W + f] += h[src[e]*W + f]   (z pre-initialized to h)
// One thread per (edge, 4-feature quad): float4 load + 4 f32 atomics.
// widthShift = log2(W/4): 4 for W=64, 5 for W=128.
// ---------------------------------------------------------------------------
__global__ void gin_scatter_add(const float* __restrict__ h,
                                const int* __restrict__ src,
                                const int* __restrict__ dst,
                                float* __restrict__ z,
                                unsigned nEdges, int qShift) {
    unsigned idx = blockIdx.x * blockDim.x + threadIdx.x;
    unsigned total = nEdges << qShift;
    if (idx >= total) return;
    unsigned e = idx >> qShift;
    unsigned q = idx & ((1u << qShift) - 1u);
    int s = src[e];
    int d = dst[e];
    const float4* hv = (const float4*)(h + (((size_t)s << qShift) + q) * 4);
    float4 v = *hv;
    float* zp = z + (((size_t)d << qShift) + q) * 4;
    atomicAdd(zp + 0, v.x);
    atomicAdd(zp + 1, v.y);
    atomicAdd(zp + 2, v.z);
    atomicAdd(zp + 3, v.w);
}

// ---------------------------------------------------------------------------
// WMMA fp32 GEMM: out(N x 64) = A(N x K) @ W(K x 64) + bias, optional ReLU.
// One wave computes a 16x16 tile via V_WMMA_F32_16X16X4_F32 (K step 4),
// fully unrolled (K is a template constant). Weights staged transposed in
// LDS so both A and B fragments are single 64-bit loads per step.
// Block = 256 threads = 8 waves -> 32 rows x 64 cols per block.
// A-frag (ISA 16x4): lanes 0-15 hold K=k,k+1; lanes 16-31 K=k+2,k+3.
// B-frag (row-striped like C/D): v0 holds K = k + 2*half, v1 = +1.
// D layout: VGPR r -> row (r + 8*half), col = lane%16.
// Row index is CLAMPED (not predicated): garbage rows only feed unstored
// output rows, so no exec-mask games in the hot loop; EXEC stays all-ones.
// ---------------------------------------------------------------------------
template<int K>
__global__ void gemm_wmma_f32(const float* __restrict__ A,
                              const float* __restrict__ W,
                              const float* __restrict__ bias,
                              float* __restrict__ out,
                              int N, int doRelu) {
    __shared__ float WT[HID * K];   // W transposed: WT[c*K + k] = W[k*64 + c]
    for (int i = threadIdx.x; i < K * HID; i += 256) {
        int k = i >> 6;
        int c = i & 63;
        WT[c * K + k] = W[i];
    }
    __syncthreads();

    const int lane = threadIdx.x & 31;
    const int wave = threadIdx.x >> 5;
    const int half = lane >> 4;     // 0: lanes 0-15, 1: lanes 16-31
    const int lid  = lane & 15;

    const int colBase = (wave & 3) * 16;                 // 4 col tiles cover 64
    const int rowBase = blockIdx.x * 32 + (wave >> 2) * 16;

    int arow = rowBase + lid;                            // M for A fragment
    arow = arow < N ? arow : (N - 1);                    // clamp, no predication
    const int col = colBase + lid;                       // N for B/D fragment

    const float* __restrict__ Arow = A + (size_t)arow * K + half * 2;
    const float* __restrict__ Bcol = WT + col * K + half * 2;

    v8f acc = {};
#pragma unroll
    for (int k = 0; k < K; k += 4) {
        v2f a = *(const v2f*)(Arow + k);   // global_load_b64
        v2f b = *(const v2f*)(Bcol + k);   // ds_load_b64
        acc = __builtin_amdgcn_wmma_f32_16x16x4_f32(
            /*neg_a=*/false, a, /*neg_b=*/false, b,
            /*c_mod=*/(short)0, acc, /*reuse_a=*/false, /*reuse_b=*/false);
    }

    float bv = bias ? bias[col] : 0.0f;
#pragma unroll
    for (int r = 0; r < 8; ++r) {
        int orow = rowBase + r + half * 8;
        if (orow < N) {
            float v = acc[r] + bv;
            if (doRelu) v = fmaxf(v, 0.0f);
            out[(size_t)orow * HID + col] = v;
        }
    }
}

// ---------------------------------------------------------------------------
// BatchNorm stats: accum[c] += sum over rows, accum[64+c] += sum of squares.
// 256 threads: 64 columns x 4 row-groups, grid-stride over rows.
// ---------------------------------------------------------------------------
__global__ void bn_stats(const float* __restrict__ t,
                         float* __restrict__ accum, int N) {
    const int c = threadIdx.x & 63;
    const int rg = threadIdx.x >> 6;   // 0..3
    float s = 0.0f, s2 = 0.0f;
    for (int r = blockIdx.x * 4 + rg; r < N; r += gridDim.x * 4) {
        float v = t[(size_t)r * HID + c];
        s += v;
        s2 += v * v;
    }
    __shared__ float ls[256], lq[256];
    ls[threadIdx.x] = s;
    lq[threadIdx.x] = s2;
    __syncthreads();
    if (rg == 0) {
        s  = ls[c] + ls[64 + c] + ls[128 + c] + ls[192 + c];
        s2 = lq[c] + lq[64 + c] + lq[128 + c] + lq[192 + c];
        atomicAdd(&accum[c], s);
        atomicAdd(&accum[64 + c], s2);
    }
}

// scale = g * rsqrt(var + eps); shift = beta - mu*scale  (biased variance)
__global__ void bn_finalize(const float* __restrict__ accum,
                            const float* __restrict__ g,
                            const float* __restrict__ be,
                            float* __restrict__ ss, float invN) {
    int c = threadIdx.x;   // 64 threads
    float mu  = accum[c] * invN;
    float var = accum[64 + c] * invN - mu * mu;
    float sc  = g[c] * rsqrtf(var + BN_EPS);
    ss[c] = sc;
    ss[64 + c] = be[c] - mu * sc;
}

__global__ void bn_apply_relu(float* __restrict__ t,
                              const float* __restrict__ ss, int N) {
    unsigned idx = blockIdx.x * blockDim.x + threadIdx.x;
    if (idx >= (unsigned)N * HID) return;
    int c = idx & 63;
    float v = fmaf(t[idx], ss[c], ss[64 + c]);
    t[idx] = fmaxf(v, 0.0f);
}

// ---------------------------------------------------------------------------
// Global mean pool
// ---------------------------------------------------------------------------
__global__ void pool_accum(const float* __restrict__ h,
                           const int* __restrict__ batch,
                           float* __restrict__ sums,
                           float* __restrict__ cnts, int N) {
    unsigned idx = blockIdx.x * blockDim.x + threadIdx.x;
    if (idx >= (unsigned)N * HID) return;
    unsigned n = idx >> 6;
    unsigned f = idx & 63;
    int b = batch[n];
    atomicAdd(&sums[(size_t)b * HID + f], h[idx]);
    if (f == 0) atomicAdd(&cnts[b], 1.0f);
}

__global__ void pool_finalize(const float* __restrict__ sums,
                              const float* __restrict__ cnts,
                              float* __restrict__ pooled) {
    unsigned idx = blockIdx.x * blockDim.x + threadIdx.x;
    if (idx >= N_GRAPHS * HID) return;
    unsigned g = idx >> 6;
    pooled[idx] = sums[idx] / fmaxf(cnts[g], 1.0f);
}

// fc2: out(512 x 10) = z(512 x 64) @ W(64 x 10) + b
__global__ void fc2_kernel(const float* __restrict__ z,
                           const float* __restrict__ W,
                           const float* __restrict__ b,
                           float* __restrict__ out) {
    int idx = blockIdx.x * blockDim.x + threadIdx.x;
    if (idx >= N_GRAPHS * OUT_DIM) return;
    int g = idx / OUT_DIM;
    int o = idx % OUT_DIM;
    float s = b[o];
#pragma unroll
    for (int k = 0; k < HID; ++k)
        s += z[g * HID + k] * W[k * OUT_DIM + o];
    out[idx] = s;
}

// ---------------------------------------------------------------------------
extern "C" void kernel_launch(void* const* d_in, const int* in_sizes, int n_in,
                              void* d_out, int out_size, void* d_ws, size_t ws_size,
                              hipStream_t stream) {
    const float* x      = (const float*)d_in[0];   // N x 128
    const int*   ei     = (const int*)  d_in[1];   // 2 x E
    const int*   batch  = (const int*)  d_in[2];   // N
    const float* W1_0   = (const float*)d_in[3];   // 128 x 64
    const float* b1_0   = (const float*)d_in[4];   // 64
    const float* W1     = (const float*)d_in[5];   // 4 x 64 x 64
    const float* b1     = (const float*)d_in[6];   // 4 x 64
    const float* W2     = (const float*)d_in[7];   // 5 x 64 x 64
    const float* b2     = (const float*)d_in[8];   // 5 x 64
    const float* g_mlp  = (const float*)d_in[9];   // 5 x 64
    const float* be_mlp = (const float*)d_in[10];  // 5 x 64
    const float* g_out  = (const float*)d_in[11];  // 4 x 64
    const float* be_out = (const float*)d_in[12];  // 4 x 64
    const float* fc1_W  = (const float*)d_in[13];  // 64 x 64
    const float* fc1_b  = (const float*)d_in[14];  // 64
    const float* fc2_W  = (const float*)d_in[15];  // 64 x 10
    const float* fc2_b  = (const float*)d_in[16];  // 10

    const int N = in_sizes[0] / IN_DIM;            // 100000
    const int E = in_sizes[1] / 2;                 // 1600000
    const int* src = ei;
    const int* dst = ei + E;

    // Workspace carve-out
    float* ws = (float*)d_ws;
    float* Z      = ws;                             // N x 128 (layer0) / N x 64
    float* T      = Z + (size_t)N * IN_DIM;         // N x 64
    float* H      = T + (size_t)N * HID;            // N x 64
    float* accum  = H + (size_t)N * HID;            // 128
    float* ss     = accum + 128;                    // 128
    float* sums   = ss + 128;                       // 512 x 64
    float* cnts   = sums + (size_t)N_GRAPHS * HID;  // 512
    float* pooled = cnts + N_GRAPHS;                // 512 x 64
    float* z1     = pooled + (size_t)N_GRAPHS * HID;// 512 x 64
    size_t needed = (size_t)(z1 + (size_t)N_GRAPHS * HID - ws) * sizeof(float);
    if (ws_size < needed) return;

    const int threads = 256;
    const int gemmBlocks = (N + 31) / 32;
    const unsigned elemsN = (unsigned)N * HID;
    const int elemBlocks = (int)((elemsN + threads - 1) / threads);

    for (int i = 0; i < 5; ++i) {
        const float* hin = (i == 0) ? x : H;
        const int K = (i == 0) ? IN_DIM : HID;
        const int qShift = (i == 0) ? 5 : 4;       // log2(K/4)
        const float* W1l = (i == 0) ? W1_0 : (W1 + (size_t)(i - 1) * HID * HID);
        const float* b1l = (i == 0) ? b1_0 : (b1 + (size_t)(i - 1) * HID);

        // z = h + sum-aggregate(h[src] -> dst)
        hipMemcpyAsync(Z, hin, (size_t)N * K * sizeof(float),
                       hipMemcpyDeviceToDevice, stream);
        {
            unsigned total = (unsigned)E << qShift;
            int blks = (int)((total + threads - 1) / threads);
            gin_scatter_add<<<blks, threads, 0, stream>>>(hin, src, dst, Z,
                                                          (unsigned)E, qShift);
        }

        // Linear 1: T = Z @ W1l + b1l
        if (i == 0)
            gemm_wmma_f32<IN_DIM><<<gemmBlocks, threads, 0, stream>>>(
                Z, W1l, b1l, T, N, 0);
        else
            gemm_wmma_f32<HID><<<gemmBlocks, threads, 0, stream>>>(
                Z, W1l, b1l, T, N, 0);

        // BN + ReLU (inner MLP norm)
        hipMemsetAsync(accum, 0, 128 * sizeof(float), stream);
        bn_stats<<<1024, threads, 0, stream>>>(T, accum, N);
        bn_finalize<<<1, 64, 0, stream>>>(accum, g_mlp + i * HID, be_mlp + i * HID,
                                          ss, 1.0f / (float)N);
        bn_apply_relu<<<elemBlocks, threads, 0, stream>>>(T, ss, N);

        // Linear 2: H = T @ W2[i] + b2[i]
        gemm_wmma_f32<HID><<<gemmBlocks, threads, 0, stream>>>(
            T, W2 + (size_t)i * HID * HID, b2 + (size_t)i * HID, H, N, 0);

        // Outer BN + ReLU on all but last layer
        if (i < 4) {
            hipMemsetAsync(accum, 0, 128 * sizeof(float), stream);
            bn_stats<<<1024, threads, 0, stream>>>(H, accum, N);
            bn_finalize<<<1, 64, 0, stream>>>(accum, g_out + i * HID,
                                              be_out + i * HID, ss, 1.0f / (float)N);
            bn_apply_relu<<<elemBlocks, threads, 0, stream>>>(H, ss, N);
        }
    }

    // Global mean pool
    hipMemsetAsync(sums, 0, (size_t)(N_GRAPHS * HID + N_GRAPHS) * sizeof(float),
                   stream);
    pool_accum<<<elemBlocks, threads, 0, stream>>>(H, batch, sums, cnts, N);
    pool_finalize<<<(N_GRAPHS * HID + threads - 1) / threads, threads, 0, stream>>>(
        sums, cnts, pooled);

    // fc1 (WMMA, ReLU fused) then fc2
    gemm_wmma_f32<HID><<<(N_GRAPHS + 31) / 32, threads, 0, stream>>>(
        pooled, fc1_W, fc1_b, z1, N_GRAPHS, 1);
    fc2_kernel<<<(N_GRAPHS * OUT_DIM + threads - 1) / threads, threads, 0, stream>>>(
        z1, fc2_W, fc2_b, (float*)d_out);
}